// ODEBlock_10746008174806
// MI455X (gfx1250) — compile-verified
//
#include <hip/hip_runtime.h>

// Neural-ODE dopri5 (B=32768, D=64, H=128, 64 bounded iterations).
// fp32 WMMA (v_wmma_f32_16x16x4_f32) for all GEMMs; working set (~90MB)
// stays L2-resident (192MB). Weights pre-packed into float2 pairs so each
// B fragment is a single global_load_b64. Requires ws_size >= ~102 MB.

#define NB     32768
#define DD     64
#define HH     128
#define BD     (NB * DD)     // 2,097,152
#define BHN    (NB * HH)     // 4,194,304
#define NITER  64

typedef __attribute__((ext_vector_type(2))) float v2f;
typedef __attribute__((ext_vector_type(8))) float v8f;

#define WMMA_F32(A_, B_, C_) \
    __builtin_amdgcn_wmma_f32_16x16x4_f32(false, (A_), false, (B_), (short)0, (C_), false, false)

// ---------------------------------------------------------------------------
// Pack W[K,N] row-major -> Wp[p][n] = (W[2p][n], W[2p+1][n]) as float2 so a
// lane's 4x16 B fragment VGPR pair is one aligned 8-byte load.
// ---------------------------------------------------------------------------
__global__ __launch_bounds__(256) void pack_w_kernel(
    const float* __restrict__ W, v2f* __restrict__ Wp, int N)
{
    const int idx = blockIdx.x * blockDim.x + threadIdx.x;   // over (K/2)*N
    const int p = idx / N;
    const int n = idx - p * N;
    v2f w;
    w.x = W[(size_t)(2 * p) * N + n];
    w.y = W[(size_t)(2 * p + 1) * N + n];
    Wp[idx] = w;
}

// ---------------------------------------------------------------------------
// fp32 WMMA GEMM:  C[M,N] = act(A[M,K] @ W[K,N] + bias)
// Block = 256 threads = 8 waves; wave tile = 32 rows x 64 cols (8 WMMA tiles).
// Per k-step (K advances by 4): 2 A b64 loads + 4 packed-B b64 loads -> 8 WMMA.
// Grid = (M/256, N/64); all tiles full -> EXEC all ones (WMMA requirement).
// A-frag (16x4 f32): lane L = row M%16; VGPR pair = K{0,1} (lanes 0-15) or
// K{2,3} (lanes 16-31).  B-frag (4x16): VGPR0=K{0|2}, VGPR1=K{1|3}, N=L%16.
// C/D (16x16 f32): VGPR i -> M = i + 8*(L/16), N = L%16.
// ---------------------------------------------------------------------------
template <int K, int N, bool RELU>
__global__ __launch_bounds__(256) void gemm_wmma_kernel(
    const float* __restrict__ A, const v2f* __restrict__ Wp,
    const float* __restrict__ bias, float* __restrict__ C)
{
    const int lane = threadIdx.x & 31;
    const int wave = threadIdx.x >> 5;
    const int half = lane >> 4;    // 0 or 1
    const int l16  = lane & 15;
    const int row0 = blockIdx.x * 256 + wave * 32;
    const int col0 = blockIdx.y * 64;
    const int n0   = col0 + l16;

    v8f a00 = {}; v8f a01 = {}; v8f a02 = {}; v8f a03 = {};
    v8f a10 = {}; v8f a11 = {}; v8f a12 = {}; v8f a13 = {};
    const float* Arow0 = A + (size_t)(row0 + l16) * K;
    const float* Arow1 = Arow0 + (size_t)16 * K;

#pragma unroll
    for (int k = 0; k < K; k += 4) {
        const int ka = k + 2 * half;                    // even -> 8B aligned
        v2f fa0 = *(const v2f*)(Arow0 + ka);
        v2f fa1 = *(const v2f*)(Arow1 + ka);
        const v2f* Wrow = Wp + (size_t)(k / 2 + half) * N;
        v2f b0 = Wrow[n0];
        v2f b1 = Wrow[n0 + 16];
        v2f b2 = Wrow[n0 + 32];
        v2f b3 = Wrow[n0 + 48];
        a00 = WMMA_F32(fa0, b0, a00);
        a01 = WMMA_F32(fa0, b1, a01);
        a02 = WMMA_F32(fa0, b2, a02);
        a03 = WMMA_F32(fa0, b3, a03);
        a10 = WMMA_F32(fa1, b0, a10);
        a11 = WMMA_F32(fa1, b1, a11);
        a12 = WMMA_F32(fa1, b2, a12);
        a13 = WMMA_F32(fa1, b3, a13);
    }

#pragma unroll
    for (int r = 0; r < 2; ++r) {
        const int rbase = row0 + 16 * r + 8 * half;
#pragma unroll
        for (int t = 0; t < 4; ++t) {
            v8f acc = (r == 0)
                ? ((t == 0) ? a00 : (t == 1) ? a01 : (t == 2) ? a02 : a03)
                : ((t == 0) ? a10 : (t == 1) ? a11 : (t == 2) ? a12 : a13);
            const int col = n0 + t * 16;
            const float bv = bias[col];
#pragma unroll
            for (int i = 0; i < 8; ++i) {
                float v = acc[i] + bv;
                if (RELU) v = fmaxf(v, 0.0f);
                C[(size_t)(rbase + i) * N + col] = v;
            }
        }
    }
}

// ---------------------------------------------------------------------------
// Elementwise: dst = y + hs * (c1*k1 + c2*k2 + c3*k3 + c4*k4 + c5*k5)
// hs read from device scalar S[3] (set per iteration by prestep).
// ---------------------------------------------------------------------------
__global__ __launch_bounds__(256) void combine_kernel(
    float* __restrict__ dst, const float* __restrict__ y,
    const float* __restrict__ S,
    const float* __restrict__ k1, const float* __restrict__ k2,
    const float* __restrict__ k3, const float* __restrict__ k4,
    const float* __restrict__ k5,
    float c1, float c2, float c3, float c4, float c5)
{
    const float hs = S[3];
    const int i = blockIdx.x * blockDim.x + threadIdx.x;
    float s = c1 * k1[i] + c2 * k2[i] + c3 * k3[i] + c4 * k4[i] + c5 * k5[i];
    dst[i] = y[i] + hs * s;
}

// err_norm^2 partial reduction:  sum over ((hs * dot(E,k)) / (atol+rtol*max|y|))^2
__global__ __launch_bounds__(256) void err_reduce_kernel(
    const float* __restrict__ y, const float* __restrict__ y5,
    float* __restrict__ S,
    const float* __restrict__ k1, const float* __restrict__ k3,
    const float* __restrict__ k4, const float* __restrict__ k5,
    const float* __restrict__ k6, const float* __restrict__ k7,
    float e0, float e2, float e3, float e4, float e5, float e6, int n)
{
    __shared__ float red[256];
    const float hs = S[3];
    float local = 0.0f;
    for (int i = blockIdx.x * blockDim.x + threadIdx.x; i < n;
         i += gridDim.x * blockDim.x) {
        float err = hs * (e0 * k1[i] + e2 * k3[i] + e3 * k4[i] +
                          e4 * k5[i] + e5 * k6[i] + e6 * k7[i]);
        float sc = 1e-3f + 1e-3f * fmaxf(fabsf(y[i]), fabsf(y5[i]));
        float r = err / sc;
        local += r * r;
    }
    red[threadIdx.x] = local;
    __syncthreads();
    for (int s = 128; s > 0; s >>= 1) {
        if (threadIdx.x < s) red[threadIdx.x] += red[threadIdx.x + s];
        __syncthreads();
    }
    if (threadIdx.x == 0) atomicAdd(&S[4], red[0]);
}

// Scalar state: S[0]=t S[1]=h S[2]=done S[3]=hs S[4]=err2sum S[5]=accept
__global__ void init_state_kernel(float* S)
{
    S[0] = 0.0f; S[1] = 0.05f; S[2] = 0.0f; S[3] = 0.0f; S[4] = 0.0f; S[5] = 0.0f;
}

__global__ void prestep_kernel(float* S)
{
    S[3] = fminf(S[1], 1.0f - S[0]);   // hs = min(h, 1 - t)
    S[4] = 0.0f;                        // reset err^2 accumulator
}

__global__ void finalize_kernel(float* S, float nelem)
{
    const float err_norm = sqrtf(S[4] / nelem);
    const float done = S[2];
    const bool accept = (err_norm <= 1.0f) && (done == 0.0f);
    const float en = fmaxf(err_norm, 1e-10f);
    float factor = 0.9f * powf(en, -0.2f);
    factor = fminf(fmaxf(factor, 0.2f), 10.0f);
    const float hs = S[3];
    const float t_new = accept ? (S[0] + hs) : S[0];
    const float h_new = (done != 0.0f) ? S[1] : hs * factor;
    S[0] = t_new;
    S[1] = h_new;
    S[5] = accept ? 1.0f : 0.0f;
    S[2] = (done != 0.0f || t_new >= 1.0f - 1e-7f) ? 1.0f : 0.0f;
}

__global__ __launch_bounds__(256) void yupdate_kernel(
    float* __restrict__ y, const float* __restrict__ y5,
    const float* __restrict__ S)
{
    if (S[5] != 0.0f) {
        const int i = blockIdx.x * blockDim.x + threadIdx.x;
        y[i] = y5[i];
    }
}

__global__ __launch_bounds__(256) void copy_kernel(
    float* __restrict__ dst, const float* __restrict__ src)
{
    const int i = blockIdx.x * blockDim.x + threadIdx.x;
    dst[i] = src[i];
}

// ---------------------------------------------------------------------------
extern "C" void kernel_launch(void* const* d_in, const int* in_sizes, int n_in,
                              void* d_out, int out_size, void* d_ws, size_t ws_size,
                              hipStream_t stream)
{
    const float* x  = (const float*)d_in[0];
    const float* W1 = (const float*)d_in[1];
    const float* b1 = (const float*)d_in[2];
    const float* W2 = (const float*)d_in[3];
    const float* b2 = (const float*)d_in[4];
    const float* W3 = (const float*)d_in[5];
    const float* b3 = (const float*)d_in[6];
    float* y = (float*)d_out;   // live ODE state; final value is the answer

    float* S    = (float*)d_ws;       // 64 scalar slots
    float* y5   = S + 64;             // BD
    float* ytmp = y5 + BD;            // BD; reused as k7 after stage combos
    float* h1   = ytmp + BD;          // BHN
    float* h2   = h1 + BHN;           // BHN
    float* K1   = h2 + BHN;           // 6 * BD
    float* K2 = K1 + BD;  float* K3 = K2 + BD;  float* K4 = K3 + BD;
    float* K5 = K4 + BD;  float* K6 = K5 + BD;
    // Packed weights (float2 pairs): sizes (K/2)*N float2 each.
    v2f* W1p = (v2f*)(K6 + BD);                 // (64/2)*128  = 4096
    v2f* W2p = W1p + (DD / 2) * HH;             // (128/2)*128 = 8192
    v2f* W3p = W2p + (HH / 2) * HH;             // (128/2)*64  = 4096

    // dopri5 tableau (match reference fp32 values)
    const float a21 = (float)(1.0/5.0);
    const float a31 = (float)(3.0/40.0),        a32 = (float)(9.0/40.0);
    const float a41 = (float)(44.0/45.0),       a42 = (float)(-56.0/15.0),     a43 = (float)(32.0/9.0);
    const float a51 = (float)(19372.0/6561.0),  a52 = (float)(-25360.0/2187.0);
    const float a53 = (float)(64448.0/6561.0),  a54 = (float)(-212.0/729.0);
    const float a61 = (float)(9017.0/3168.0),   a62 = (float)(-355.0/33.0);
    const float a63 = (float)(46732.0/5247.0),  a64 = (float)(49.0/176.0),     a65 = (float)(-5103.0/18656.0);
    const float c1f = (float)(35.0/384.0), c3f = (float)(500.0/1113.0), c4f = (float)(125.0/192.0);
    const float c5f = (float)(-2187.0/6784.0), c6f = (float)(11.0/84.0);
    const float E0 = c1f - (float)(5179.0/57600.0);
    const float E2 = c3f - (float)(7571.0/16695.0);
    const float E3 = c4f - (float)(393.0/640.0);
    const float E4 = c5f - (float)(-92097.0/339200.0);
    const float E5 = c6f - (float)(187.0/2100.0);
    const float E6 = -(float)(1.0/40.0);

    const dim3 blk(256);
    const dim3 ew(BD / 256);                    // elementwise over y-shaped arrays
    const dim3 g1(NB / 256, HH / 64);           // fc1: [B,64]x[64,128]
    const dim3 g2(NB / 256, HH / 64);           // fc2: [B,128]x[128,128]
    const dim3 g3(NB / 256, DD / 64);           // fc3: [B,128]x[128,64]

    // One-time (per launch) weight packing + state init.
    pack_w_kernel<<<(DD / 2) * HH / 256, blk, 0, stream>>>(W1, W1p, HH);
    pack_w_kernel<<<(HH / 2) * HH / 256, blk, 0, stream>>>(W2, W2p, HH);
    pack_w_kernel<<<(HH / 2) * DD / 256, blk, 0, stream>>>(W3, W3p, DD);
    copy_kernel<<<ew, blk, 0, stream>>>(y, x);
    init_state_kernel<<<1, 1, 0, stream>>>(S);

    auto feval = [&](const float* src, float* kdst) {
        gemm_wmma_kernel<DD, HH, true ><<<g1, blk, 0, stream>>>(src, W1p, b1, h1);
        gemm_wmma_kernel<HH, HH, true ><<<g2, blk, 0, stream>>>(h1,  W2p, b2, h2);
        gemm_wmma_kernel<HH, DD, false><<<g3, blk, 0, stream>>>(h2,  W3p, b3, kdst);
    };

    for (int it = 0; it < NITER; ++it) {
        prestep_kernel<<<1, 1, 0, stream>>>(S);

        feval(y, K1);                                                     // k1 = f(y)
        combine_kernel<<<ew, blk, 0, stream>>>(ytmp, y, S, K1, K1, K1, K1, K1,
                                               a21, 0.f, 0.f, 0.f, 0.f);
        feval(ytmp, K2);
        combine_kernel<<<ew, blk, 0, stream>>>(ytmp, y, S, K1, K2, K1, K1, K1,
                                               a31, a32, 0.f, 0.f, 0.f);
        feval(ytmp, K3);
        combine_kernel<<<ew, blk, 0, stream>>>(ytmp, y, S, K1, K2, K3, K1, K1,
                                               a41, a42, a43, 0.f, 0.f);
        feval(ytmp, K4);
        combine_kernel<<<ew, blk, 0, stream>>>(ytmp, y, S, K1, K2, K3, K4, K1,
                                               a51, a52, a53, a54, 0.f);
        feval(ytmp, K5);
        combine_kernel<<<ew, blk, 0, stream>>>(ytmp, y, S, K1, K2, K3, K4, K5,
                                               a61, a62, a63, a64, a65);
        feval(ytmp, K6);

        // 5th-order solution y5 = y + hs*(b5 . k)
        combine_kernel<<<ew, blk, 0, stream>>>(y5, y, S, K1, K3, K4, K5, K6,
                                               c1f, c3f, c4f, c5f, c6f);
        feval(y5, ytmp);                                                  // k7 -> ytmp

        err_reduce_kernel<<<2048, blk, 0, stream>>>(y, y5, S, K1, K3, K4, K5, K6,
                                                    ytmp, E0, E2, E3, E4, E5, E6, BD);
        finalize_kernel<<<1, 1, 0, stream>>>(S, (float)BD);
        yupdate_kernel<<<ew, blk, 0, stream>>>(y, y5, S);
    }
}